// TripletLoss_28338194219418
// MI455X (gfx1250) — compile-verified
//
#include <hip/hip_runtime.h>
#include <hip/hip_bf16.h>

typedef __attribute__((ext_vector_type(2))) float v2f;
typedef __attribute__((ext_vector_type(8))) float v8f;

#define NROW 8192
#define DDIM 512
#define LDSTR 516   // 512 + 4 pad: lanes 0..15 hit banks 4r, lanes 16..31 banks 4r+2 -> conflict-free
#define MARGIN 0.5f
#define EPSD 1e-12f

// ---------------- prep: x = inputs * A, sq[r] = sum(x_r^2) ----------------
__global__ __launch_bounds__(256) void prep_kernel(const float* __restrict__ inp,
                                                   const float* __restrict__ aw,
                                                   float* __restrict__ x,
                                                   float* __restrict__ sq) {
    const int r = blockIdx.x;
    const int t = threadIdx.x;
    const size_t base = (size_t)r * DDIM;
    const int k = 2 * t;                      // 256 threads x 2 elems = 512
    float x0 = inp[base + k]     * aw[base + k];
    float x1 = inp[base + k + 1] * aw[base + k + 1];
    x[base + k]     = x0;
    x[base + k + 1] = x1;
    float s = x0 * x0 + x1 * x1;
    #pragma unroll
    for (int m = 1; m < 32; m <<= 1) s += __shfl_xor(s, m, 32);
    __shared__ float wsum[8];
    if ((t & 31) == 0) wsum[t >> 5] = s;
    __syncthreads();
    if (t == 0) {
        float tot = 0.f;
        #pragma unroll
        for (int i = 0; i < 8; ++i) tot += wsum[i];
        sq[r] = tot;
    }
}

// ---------------- init ap/an (int-bit encoded nonneg floats) ----------------
__global__ __launch_bounds__(256) void init_kernel(int* __restrict__ apb, int* __restrict__ anb) {
    int i = blockIdx.x * 256 + threadIdx.x;
    if (i < NROW) { apb[i] = 0; anb[i] = 0x7F800000; }  // 0.0f and +inf
}

// ---------------- fused Gram + distance + hard mining ----------------
// grid = (NROW/16, COLCHUNKS); block = 256 (8 waves). Each wave: 16x64 col blocks,
// 4 f32-WMMA accumulators, K in steps of 4.
__global__ __launch_bounds__(256) void tile_kernel(const float* __restrict__ x,
                                                   const float* __restrict__ sq,
                                                   const int* __restrict__ tgt,
                                                   int* __restrict__ apb,
                                                   int* __restrict__ anb) {
    __shared__ float As[16 * LDSTR];
    __shared__ float sqr[16];
    __shared__ int   trow[16];

    const int rowbase = blockIdx.x * 16;
    const int tid = threadIdx.x;

    // stage the 16-row A strip into LDS (coalesced)
    for (int i = tid; i < 16 * DDIM; i += 256) {
        int r = i >> 9;           // /512
        int k = i & (DDIM - 1);
        As[r * LDSTR + k] = x[(size_t)(rowbase + r) * DDIM + k];
    }
    if (tid < 16) { sqr[tid] = sq[rowbase + tid]; trow[tid] = tgt[rowbase + tid]; }
    __syncthreads();

    const int wave = tid >> 5;
    const int lane = tid & 31;
    const int h    = lane >> 4;   // K-pair half: lanes 0-15 hold K=0,1 ; 16-31 hold K=2,3
    const int ln   = lane & 15;   // M (A-frag) / N (B-, C-frag) index

    float apv[8], anv[8];
    #pragma unroll
    for (int v = 0; v < 8; ++v) { apv[v] = 0.0f; anv[v] = __int_as_float(0x7F800000); }

    const int colsPerChunk  = NROW / gridDim.y;
    const int blocksPerChunk = colsPerChunk >> 6;          // 64 cols per block
    const int chunkbase = blockIdx.y * colsPerChunk;

    const float* __restrict__ xr = &As[ln * LDSTR + 2 * h];

    for (int cb = wave; cb < blocksPerChunk; cb += 8) {
        const int colbase = chunkbase + (cb << 6);
        const float* __restrict__ xc0 = &x[(size_t)(colbase + ln) * DDIM + 2 * h];
        const float* __restrict__ xc1 = xc0 + 16 * DDIM;
        const float* __restrict__ xc2 = xc0 + 32 * DDIM;
        const float* __restrict__ xc3 = xc0 + 48 * DDIM;

        v8f acc0 = {}, acc1 = {}, acc2 = {}, acc3 = {};

        #pragma unroll 8
        for (int k0 = 0; k0 < DDIM; k0 += 4) {
            v2f a  = *(const v2f*)(xr  + k0);   // LDS, bank-conflict-free
            v2f b0 = *(const v2f*)(xc0 + k0);   // global, L2-resident (x = 16MB << 192MB L2)
            v2f b1 = *(const v2f*)(xc1 + k0);
            v2f b2 = *(const v2f*)(xc2 + k0);
            v2f b3 = *(const v2f*)(xc3 + k0);
            acc0 = __builtin_amdgcn_wmma_f32_16x16x4_f32(false, a, false, b0, (short)0, acc0, false, false);
            acc1 = __builtin_amdgcn_wmma_f32_16x16x4_f32(false, a, false, b1, (short)0, acc1, false, false);
            acc2 = __builtin_amdgcn_wmma_f32_16x16x4_f32(false, a, false, b2, (short)0, acc2, false, false);
            acc3 = __builtin_amdgcn_wmma_f32_16x16x4_f32(false, a, false, b3, (short)0, acc3, false, false);
        }

        // fused epilogue: dist = sqrt(clip(sq_i + sq_j - 2 G, eps)), masked hard mining
        #pragma unroll
        for (int t = 0; t < 4; ++t) {
            v8f acc = (t == 0) ? acc0 : (t == 1) ? acc1 : (t == 2) ? acc2 : acc3;
            const int col = colbase + (t << 4) + ln;   // C/D layout: N = lane%16
            const float sqc = sq[col];
            const int   tc  = tgt[col];
            #pragma unroll
            for (int v = 0; v < 8; ++v) {
                const int m = v + 8 * h;               // C/D layout: M = v + 8*(lane/16)
                float d2   = sqr[m] + sqc - 2.0f * acc[v];
                float dist = __builtin_sqrtf(fmaxf(d2, EPSD));
                bool same  = (trow[m] == tc);
                apv[v] = (same  && dist > apv[v]) ? dist : apv[v];
                anv[v] = (!same && dist < anv[v]) ? dist : anv[v];
            }
        }
    }

    // reduce over the 16 N-lanes within each half-wave, then one atomic per row
    #pragma unroll
    for (int v = 0; v < 8; ++v) {
        float ap = apv[v], an = anv[v];
        #pragma unroll
        for (int m = 1; m < 16; m <<= 1) {
            ap = fmaxf(ap, __shfl_xor(ap, m, 32));
            an = fminf(an, __shfl_xor(an, m, 32));
        }
        if (ln == 0) {
            const int row = rowbase + v + 8 * h;
            atomicMax(&apb[row], __float_as_int(ap));  // nonneg float: int order == float order
            atomicMin(&anb[row], __float_as_int(an));
        }
    }
}

// ---------------- final: mean(relu(ap - an + margin)) ----------------
__global__ __launch_bounds__(256) void final_kernel(const int* __restrict__ apb,
                                                    const int* __restrict__ anb,
                                                    float* __restrict__ out) {
    float s = 0.f;
    for (int r = threadIdx.x; r < NROW; r += 256) {
        float ap = __int_as_float(apb[r]);
        float an = __int_as_float(anb[r]);
        s += fmaxf(ap - an + MARGIN, 0.0f);
    }
    #pragma unroll
    for (int m = 1; m < 32; m <<= 1) s += __shfl_xor(s, m, 32);
    __shared__ float wsum[8];
    if ((threadIdx.x & 31) == 0) wsum[threadIdx.x >> 5] = s;
    __syncthreads();
    if (threadIdx.x == 0) {
        float tot = 0.f;
        #pragma unroll
        for (int i = 0; i < 8; ++i) tot += wsum[i];
        out[0] = tot / (float)NROW;
    }
}

extern "C" void kernel_launch(void* const* d_in, const int* in_sizes, int n_in,
                              void* d_out, int out_size, void* d_ws, size_t ws_size,
                              hipStream_t stream) {
    const float* inp = (const float*)d_in[0];   // [8192, 512] f32
    const float* aw  = (const float*)d_in[1];   // [8192, 512] f32
    const int*   tgt = (const int*)d_in[2];     // [8192] int

    // workspace: x (16 MB) | sq (32 KB) | ap (32 KB) | an (32 KB)
    float* x  = (float*)d_ws;
    float* sq = x + (size_t)NROW * DDIM;
    int*   apb = (int*)(sq + NROW);
    int*   anb = apb + NROW;

    prep_kernel<<<NROW, 256, 0, stream>>>(inp, aw, x, sq);
    init_kernel<<<NROW / 256, 256, 0, stream>>>(apb, anb);
    tile_kernel<<<dim3(NROW / 16, 4), 256, 0, stream>>>(x, sq, tgt, apb, anb);
    final_kernel<<<1, 256, 0, stream>>>(apb, anb, (float*)d_out);
}